// IFOPooling_48404281426015
// MI455X (gfx1250) — compile-verified
//
#include <hip/hip_runtime.h>
#include <stdint.h>

// Problem constants (from reference): B=16, H=1024, S=2048
#define BDIM 16
#define HDIM 1024
#define SDIM 2048
#define CHANNELS (BDIM * HDIM)     // 16384 independent scan chains
#define WAVES_PER_BLOCK 8
#define LANES 32
#define EPL 8                       // elements per lane per chunk
#define CHUNK (LANES * EPL)         // 256 floats = 1KB per array per chunk
#define NCHUNKS (SDIM / CHUNK)      // 8

typedef float v4f __attribute__((ext_vector_type(4), aligned(16)));

__device__ __forceinline__ float bperm_f(int src_lane, float v) {
  // ds_bpermute_b32: index in bytes; wave32 uses index bits [6:2] (wraps)
  return __int_as_float(
      __builtin_amdgcn_ds_bpermute(src_lane << 2, __float_as_int(v)));
}

__device__ __forceinline__ float readlane_f(float v, int lane) {
  return __int_as_float(__builtin_amdgcn_readlane(__float_as_int(v), lane));
}

__global__ __launch_bounds__(256)
void ifo_scan_kernel(const float* __restrict__ f,
                     const float* __restrict__ z,
                     const float* __restrict__ i_,
                     float* __restrict__ out) {
  // LDS staging: [wave][double-buffer][f/z/i][CHUNK floats] = 48KB
  __shared__ __align__(16) float lds[WAVES_PER_BLOCK][2][3][CHUNK];

  const int tid  = threadIdx.x;
  const int wave = tid >> 5;
  const int lane = tid & 31;
  const int ch   = blockIdx.x * WAVES_PER_BLOCK + wave;

  const size_t row = (size_t)ch * SDIM;
  const float* fp = f  + row;
  const float* zp = z  + row;
  const float* ip = i_ + row;
  float*       op = out + row;

  // LDS byte offsets for this lane (low 32 bits of the generic pointer
  // to a __shared__ object are the LDS offset: flat = aperture | offset).
  unsigned ldsF[2], ldsZ[2], ldsI[2];
#pragma unroll
  for (int b = 0; b < 2; ++b) {
    ldsF[b] = (unsigned)(uintptr_t)&lds[wave][b][0][lane * EPL];
    ldsZ[b] = (unsigned)(uintptr_t)&lds[wave][b][1][lane * EPL];
    ldsI[b] = (unsigned)(uintptr_t)&lds[wave][b][2][lane * EPL];
  }

  // Issue one chunk's 6 async b128 copies (global -> LDS), ASYNCcnt += 6.
  // INST_OFFSET is added to BOTH the LDS and global addresses, so one vdst
  // operand covers both 16B halves of the lane's 32B slice. Inputs are
  // streamed exactly once -> non-temporal hint keeps them out of L2.
  auto issue_chunk = [&](int k) {
    const int buf = k & 1;
    const unsigned goff =
        (unsigned)((k * CHUNK + lane * EPL) * sizeof(float));
    unsigned long long ga_f = (unsigned long long)(uintptr_t)fp + goff;
    unsigned long long ga_z = (unsigned long long)(uintptr_t)zp + goff;
    unsigned long long ga_i = (unsigned long long)(uintptr_t)ip + goff;
    // Architectural WAR guard: make sure prior ds_loads of the buffer we are
    // about to overwrite have fully executed before the async engine can
    // write it (nearly free: those reads were consumed last iteration).
    asm volatile("s_wait_dscnt 0x0" ::: "memory");
    asm volatile(
        "global_load_async_to_lds_b128 %0, %1, off th:TH_LOAD_NT\n\t"
        "global_load_async_to_lds_b128 %0, %1, off offset:16 th:TH_LOAD_NT"
        :: "v"(ldsF[buf]), "v"(ga_f) : "memory");
    asm volatile(
        "global_load_async_to_lds_b128 %0, %1, off th:TH_LOAD_NT\n\t"
        "global_load_async_to_lds_b128 %0, %1, off offset:16 th:TH_LOAD_NT"
        :: "v"(ldsZ[buf]), "v"(ga_z) : "memory");
    asm volatile(
        "global_load_async_to_lds_b128 %0, %1, off th:TH_LOAD_NT\n\t"
        "global_load_async_to_lds_b128 %0, %1, off offset:16 th:TH_LOAD_NT"
        :: "v"(ldsI[buf]), "v"(ga_i) : "memory");
  };

  issue_chunk(0);

  float hrun = 0.0f;  // h value entering the current chunk (uniform per wave)

#pragma unroll
  for (int k = 0; k < NCHUNKS; ++k) {
    if (k + 1 < NCHUNKS) {
      issue_chunk(k + 1);
      // 12 outstanding; loads complete in order, so <=6 means chunk k landed.
      asm volatile("s_wait_asynccnt 0x6" ::: "memory");
    } else {
      asm volatile("s_wait_asynccnt 0x0" ::: "memory");
    }

    const int buf = k & 1;
    const v4f* pf = (const v4f*)&lds[wave][buf][0][lane * EPL];
    const v4f* pz = (const v4f*)&lds[wave][buf][1][lane * EPL];
    const v4f* pi = (const v4f*)&lds[wave][buf][2][lane * EPL];

    float fv[EPL], xv[EPL];
    {
      v4f F0 = pf[0], F1 = pf[1];
      v4f Z0 = pz[0], Z1 = pz[1];
      v4f I0 = pi[0], I1 = pi[1];
#pragma unroll
      for (int j = 0; j < 4; ++j) {
        fv[j]     = F0[j];
        fv[4 + j] = F1[j];
        xv[j]     = Z0[j] * I0[j];
        xv[4 + j] = Z1[j] * I1[j];
      }
    }

    // Lane-local affine map over its 8 elements: h_out = A*h_in + B
    float A  = fv[0];
    float Bv = xv[0];
#pragma unroll
    for (int j = 1; j < EPL; ++j) {
      Bv = fmaf(fv[j], Bv, xv[j]);
      A  = A * fv[j];
    }

    // Wave-wide inclusive scan of affine maps (compose older-then-newer):
    //   A_i <- A_i * A_{i-d};  B_i <- A_i * B_{i-d} + B_i
#pragma unroll
    for (int d = 1; d < LANES; d <<= 1) {
      float Au = bperm_f(lane - d, A);
      float Bu = bperm_f(lane - d, Bv);
      if (lane >= d) {
        Bv = fmaf(A, Bu, Bv);
        A  = A * Au;
      }
    }

    // Exclusive carry for this lane = inclusive of lane-1 applied to hrun.
    float Ae = bperm_f(lane - 1, A);
    float Be = bperm_f(lane - 1, Bv);
    if (lane == 0) { Ae = 1.0f; Be = 0.0f; }
    float h = fmaf(Ae, hrun, Be);

    // Chunk total = lane 31 inclusive (uniform broadcast via v_readlane).
    float At = readlane_f(A, 31);
    float Bt = readlane_f(Bv, 31);
    hrun = fmaf(At, hrun, Bt);

    // Re-emit the 8 per-element h values and stream them out (non-temporal:
    // the 128MB output is never re-read, keep it out of L2).
    v4f O0, O1;
#pragma unroll
    for (int j = 0; j < 4; ++j) {
      h = fmaf(fv[j], h, xv[j]);
      O0[j] = h;
    }
#pragma unroll
    for (int j = 0; j < 4; ++j) {
      h = fmaf(fv[4 + j], h, xv[4 + j]);
      O1[j] = h;
    }
    v4f* po = (v4f*)(op + (size_t)k * CHUNK + lane * EPL);
    __builtin_nontemporal_store(O0, po);
    __builtin_nontemporal_store(O1, po + 1);
  }
}

extern "C" void kernel_launch(void* const* d_in, const int* in_sizes, int n_in,
                              void* d_out, int out_size, void* d_ws,
                              size_t ws_size, hipStream_t stream) {
  (void)in_sizes; (void)n_in; (void)out_size; (void)d_ws; (void)ws_size;
  const float* f = (const float*)d_in[0];
  const float* z = (const float*)d_in[1];
  const float* i = (const float*)d_in[2];
  float* out = (float*)d_out;

  dim3 grid(CHANNELS / WAVES_PER_BLOCK);  // 2048 workgroups
  dim3 block(WAVES_PER_BLOCK * LANES);    // 256 threads = 8 wave32
  ifo_scan_kernel<<<grid, block, 0, stream>>>(f, z, i, out);
}